// MapLoss_33423435498326
// MI455X (gfx1250) — compile-verified
//
#include <hip/hip_runtime.h>

// ---------------------------------------------------------------------------
// MapLoss for MI455X (gfx1250).
// 64 rows = 32 samples x 2 maps. One 1024-thread (32 wave32) workgroup/row.
// Exact top-k via two-stage 16-bit radix select on float bit patterns using a
// 64K-bin LDS histogram (256KB LDS -- feasible only with CDNA5's 320KB/WGP).
// Only TWO streaming passes: the second radix stage's bins are exact 32-bit
// values, so the above-threshold sum inside the selected hi-bin is
// sum(count[lo] * value) -- no third data scan needed.
// Wave reductions use V_WMMA_F32_16X16X4_F32 with an all-ones B matrix.
// ---------------------------------------------------------------------------

namespace {
constexpr int kNumB    = 32;
constexpr int kM       = 512 * 512;               // 262144 elements per row
constexpr int kThreads = 1024;                    // 32 wave32 waves
constexpr int kVec     = 4;
constexpr int kIters   = kM / (kThreads * kVec);  // 64 float4 iterations
constexpr int kBins    = 65536;
constexpr int kChunks  = 1024;
constexpr int kChunkSz = kBins / kChunks;         // 64
constexpr float    kPosThresh = 0.1f;
constexpr unsigned kTopNoPos  = 500;
}

typedef float v2f __attribute__((ext_vector_type(2)));
typedef float v8f __attribute__((ext_vector_type(8)));

// Sum of one float per lane across a wave32 via WMMA f32 16x16x4 with B=ones:
// D row m = v[m] + v[m+16]; the 8 accumulator VGPRs hold rows 0..7 (lanes<16)
// and 8..15 (lanes>=16); summing them plus one cross-half shuffle gives the
// full 32-lane sum in every lane.
__device__ __forceinline__ float wave_sum32(float v) {
#if __has_builtin(__builtin_amdgcn_wmma_f32_16x16x4_f32)
  v2f a;  a[0]  = v;    a[1]  = 0.0f;
  v2f bm; bm[0] = 1.0f; bm[1] = 1.0f;
  v8f c = {};
  c = __builtin_amdgcn_wmma_f32_16x16x4_f32(false, a, false, bm,
                                            (short)0, c, false, false);
  float s = ((c[0] + c[1]) + (c[2] + c[3])) + ((c[4] + c[5]) + (c[6] + c[7]));
  s += __shfl_xor(s, 16, 32);
  return s;
#else
  for (int off = 16; off > 0; off >>= 1) v += __shfl_xor(v, off, 32);
  return v;
#endif
}

// Deterministic block-wide sum (1024 threads = 32 waves). redf has 32 slots.
__device__ __forceinline__ float block_sum(float v, float* redf) {
  const int tid  = threadIdx.x;
  const int wave = tid >> 5;
  const int lane = tid & 31;
  float w = wave_sum32(v);
  if (lane == 0) redf[wave] = w;
  __syncthreads();
  if (wave == 0) {
    float t = redf[lane];
    t = wave_sum32(t);     // uniform per-wave branch: EXEC all 1s in wave 0
    if (lane == 0) redf[0] = t;
  }
  __syncthreads();
  float total = redf[0];
  __syncthreads();
  return total;
}

// Serial (tid 0) descending select: largest bin index `sel` such that
// suffix-count(sel) >= k. chunkTot[c] = sum of hist[c*64 .. c*64+63].
// *cntAbove = number of elements in bins strictly above sel.
__device__ __forceinline__ void select_bin(const unsigned* hist,
                                           const unsigned* chunkTot,
                                           unsigned k,
                                           unsigned* sel, unsigned* cntAbove) {
  unsigned acc = 0;
  int c = kChunks - 1;
  for (; c >= 0; --c) {
    unsigned t = chunkTot[c];
    if (acc + t >= k) break;
    acc += t;
  }
  if (c < 0) { *sel = 0; *cntAbove = acc; return; }  // k > total: unreachable
  int bbin = c * kChunkSz + kChunkSz - 1;
  for (;; --bbin) {
    unsigned h = hist[bbin];
    if (acc + h >= k) break;
    acc += h;
    if (bbin == c * kChunkSz) break;                 // safety
  }
  *sel = (unsigned)bbin;
  *cntAbove = acc;
}

__global__ void __launch_bounds__(kThreads)
row_loss_kernel(const float* __restrict__ gh_label,
                const float* __restrict__ gah_label,
                const float* __restrict__ p_gh,
                const float* __restrict__ p_gah,
                const float* __restrict__ mask,
                float* __restrict__ rowOut) {
  extern __shared__ unsigned char smem[];
  unsigned* hist     = (unsigned*)smem;              // 65536 u32 = 256 KB
  unsigned* chunkTot = hist + kBins;                 // 1024 u32
  float*    redf     = (float*)(chunkTot + kChunks); // 32 f32
  unsigned* ctl      = (unsigned*)(redf + 32);       // 8 u32 control words

  const int tid  = threadIdx.x;
  const int row  = blockIdx.x;
  const int samp = row & 31;
  const int map  = row >> 5;

  const float* lab = map ? gah_label : gh_label;
  const float* prd = map ? p_gah     : p_gh;
  const size_t base = (size_t)samp * (size_t)kM;

  const float4* lab4 = (const float4*)(lab  + base);
  const float4* prd4 = (const float4*)(prd  + base);
  const float4* msk4 = (const float4*)(mask + base);

  // ---- Pass 1: P, pos_sum, hi-16 histogram of negative losses -------------
  for (int i = tid; i < kBins; i += kThreads) hist[i] = 0;
  __syncthreads();

  float Pl = 0.0f, posSumL = 0.0f;
  for (int it = 0; it < kIters; ++it) {
    const int idx = it * kThreads + tid;
    if (it + 1 < kIters) {
      __builtin_prefetch(&lab4[idx + kThreads], 0, 1);
      __builtin_prefetch(&prd4[idx + kThreads], 0, 1);
      __builtin_prefetch(&msk4[idx + kThreads], 0, 1);
    }
    float4 l4 = lab4[idx], p4 = prd4[idx], m4 = msk4[idx];
    const float* lv = (const float*)&l4;
    const float* pv = (const float*)&p4;
    const float* mv = (const float*)&m4;
#pragma unroll
    for (int j = 0; j < kVec; ++j) {
      const float d    = pv[j] - lv[j];
      const float loss = d * d * mv[j];
      if (lv[j] >= kPosThresh) {
        Pl += 1.0f;
        posSumL += loss;
      } else {
        atomicAdd(&hist[__float_as_uint(loss) >> 16], 1u);
      }
    }
  }

  const float Pf     = block_sum(Pl, redf);
  const float posSum = block_sum(posSumL, redf);
  const unsigned P = (unsigned)(Pf + 0.5f);
  const unsigned N = (unsigned)kM - P;
  const bool isP0  = (P == 0);
  const unsigned k = isP0 ? kTopNoPos : ((3u * P < N) ? 3u * P : N);

  if (!isP0 && k == 0) {  // all pixels positive: negative term is 0
    if (tid == 0) rowOut[row] = posSum / (float)P;
    return;               // uniform across block
  }

  // ---- Select high-16 bin -------------------------------------------------
  {
    unsigned s = 0;
    const unsigned* hp = hist + tid * kChunkSz;
    for (int j = 0; j < kChunkSz; ++j) s += hp[j];
    chunkTot[tid] = s;
  }
  __syncthreads();
  if (tid == 0) {
    unsigned sel, ca;
    select_bin(hist, chunkTot, k, &sel, &ca);
    ctl[0] = sel; ctl[1] = ca;
  }
  __syncthreads();
  const unsigned selHi      = ctl[0];
  const unsigned cntAboveHi = ctl[1];

  // ---- Pass 2: lo-16 histogram inside selected hi bin, plus scalar sum of
  //      all losses whose hi-16 is strictly above the selected bin ----------
  __syncthreads();
  for (int i = tid; i < kBins; i += kThreads) hist[i] = 0;
  __syncthreads();

  float sgtL = 0.0f;  // sum over negatives with hi16 > selHi
  for (int it = 0; it < kIters; ++it) {
    const int idx = it * kThreads + tid;
    if (it + 1 < kIters) {
      __builtin_prefetch(&lab4[idx + kThreads], 0, 1);
      __builtin_prefetch(&prd4[idx + kThreads], 0, 1);
      __builtin_prefetch(&msk4[idx + kThreads], 0, 1);
    }
    float4 l4 = lab4[idx], p4 = prd4[idx], m4 = msk4[idx];
    const float* lv = (const float*)&l4;
    const float* pv = (const float*)&p4;
    const float* mv = (const float*)&m4;
#pragma unroll
    for (int j = 0; j < kVec; ++j) {
      if (lv[j] < kPosThresh) {
        const float d    = pv[j] - lv[j];
        const float loss = d * d * mv[j];
        const unsigned bits = __float_as_uint(loss);
        const unsigned hi   = bits >> 16;
        if (hi > selHi) {
          sgtL += loss;
        } else if (hi == selHi) {
          atomicAdd(&hist[bits & 0xFFFFu], 1u);
        }
      }
    }
  }
  __syncthreads();

  {
    unsigned s = 0;
    const unsigned* hp = hist + tid * kChunkSz;
    for (int j = 0; j < kChunkSz; ++j) s += hp[j];
    chunkTot[tid] = s;
  }
  __syncthreads();
  if (tid == 0) {
    unsigned sel, ca;
    select_bin(hist, chunkTot, k - cntAboveHi, &sel, &ca);
    ctl[2] = sel; ctl[3] = ca;
  }
  __syncthreads();
  const unsigned selLo  = ctl[2];
  const unsigned vkBits = (selHi << 16) | selLo;    // exact k-th largest value
  const unsigned cGt    = cntAboveHi + ctl[3];      // strictly-greater count
  const float    vk     = __uint_as_float(vkBits);

  // ---- In-bin above-threshold sum from the histogram itself ---------------
  // Every element of lo-bin `i` has the exact bit pattern (selHi<<16)|i, so
  // its contribution is count * value. No third streaming pass required.
  float saL = 0.0f;
  for (int i = tid; i < kBins; i += kThreads) {
    if ((unsigned)i > selLo) {
      const unsigned cnt = hist[i];
      if (cnt) saL += (float)cnt * __uint_as_float((selHi << 16) | (unsigned)i);
    }
  }
  const float sumGt = block_sum(sgtL + saL, redf);

  if (tid == 0) {
    const float topk = sumGt + (float)(k - cGt) * vk;  // exact tie handling
    float per;
    if (isP0) {
      per = topk / (float)kTopNoPos;
    } else {
      per = posSum / (float)P + topk / (float)k;
    }
    rowOut[row] = per;
  }
}

// Deterministic final combine: char_loss/B + affi_loss/B.
__global__ void finalize_kernel(const float* __restrict__ rowOut,
                                float* __restrict__ out) {
  if (threadIdx.x == 0 && blockIdx.x == 0) {
    float charL = 0.0f, affiL = 0.0f;
    for (int i = 0; i < kNumB; ++i) charL += rowOut[i];
    for (int i = 0; i < kNumB; ++i) affiL += rowOut[kNumB + i];
    out[0] = charL / (float)kNumB + affiL / (float)kNumB;
  }
}

extern "C" void kernel_launch(void* const* d_in, const int* in_sizes, int n_in,
                              void* d_out, int out_size, void* d_ws,
                              size_t ws_size, hipStream_t stream) {
  (void)in_sizes; (void)n_in; (void)out_size; (void)ws_size;
  const float* gh_label  = (const float*)d_in[0];
  const float* gah_label = (const float*)d_in[1];
  const float* p_gh      = (const float*)d_in[2];
  const float* p_gah     = (const float*)d_in[3];
  const float* mask      = (const float*)d_in[4];
  float* rowOut = (float*)d_ws;   // 64 floats
  float* out    = (float*)d_out;  // 1 float

  const size_t shmem = (size_t)(kBins + kChunks) * sizeof(unsigned) +
                       (size_t)(32 + 8) * sizeof(float);   // ~266.4 KB < 320 KB

  row_loss_kernel<<<2 * kNumB, kThreads, shmem, stream>>>(
      gh_label, gah_label, p_gh, p_gah, mask, rowOut);
  finalize_kernel<<<1, 64, 0, stream>>>(rowOut, out);
}